// myPointwiseConv_40140764348993
// MI455X (gfx1250) — compile-verified
//
#include <hip/hip_runtime.h>

// ---------------------------------------------------------------------------
// MI455X (gfx1250) pointwise conv with stride-2 subsample + Newton-Schulz
// orthogonalization, all matmuls on V_WMMA_F32_16X16X4_F32 (wave32).
//
// Roofline: GEMM = 13.2 GFLOP; HBM traffic ~308 MB -> ~13 us @ 23.3 TB/s.
// f32 WMMA keeps reference precision; memory-bound, so the design minimizes
// HBM traffic (x staged to LDS once, weight L2-resident) and minimizes
// hot-loop instruction count (pair-packed b64 operand fetches).
// ---------------------------------------------------------------------------

typedef __attribute__((ext_vector_type(2))) float v2f;
typedef __attribute__((ext_vector_type(8))) float v8f;

__device__ __forceinline__ v8f wmma_f32x4(v2f a, v2f b, v8f c) {
  // 8 args: (neg_a, A, neg_b, B, c_mod, C, reuse_a, reuse_b)
  return __builtin_amdgcn_wmma_f32_16x16x4_f32(false, a, false, b, (short)0, c,
                                               false, false);
}

#define C_IN   256
#define C_OUT  256
#define BATCH  32
#define H_IN   112
#define W_IN   112
#define H_OUT  56
#define W_OUT  56
#define NS_ITERS 16

// LDS B-tile: 128 k-pair rows x 32 cols, padded to 48 float2/row so that the
// two half-waves of a ds_load_b64 (adjacent k2 rows) hit disjoint bank sets
// (row offset 96 banks == 32 mod 64).
#define LDSW 48

// ---------------------------------------------------------------------------
// Newton-Schulz setup: scalar accumulator zero + Frobenius norm^2 reduction.
// ---------------------------------------------------------------------------
__global__ void k_zero(float* p) { p[0] = 0.0f; }

__global__ void k_frob(const float* __restrict__ w, float* __restrict__ norm2) {
  __shared__ float red[256];
  int t = threadIdx.x;
  float v = w[blockIdx.x * 256 + t];
  red[t] = v * v;
  __syncthreads();
  for (int s = 128; s > 0; s >>= 1) {
    if (t < s) red[t] += red[t + s];
    __syncthreads();
  }
  if (t == 0) atomicAdd(norm2, red[0]);
}

__global__ void k_scale(const float* __restrict__ w,
                        const float* __restrict__ norm2,
                        float* __restrict__ x) {
  int i = blockIdx.x * 256 + threadIdx.x;
  x[i] = w[i] * rsqrtf(*norm2 + 1e-12f);
}

// ---------------------------------------------------------------------------
// T = X^T X   (256x256x256, one wave per 16x16 tile)
// ---------------------------------------------------------------------------
__global__ void k_gram(const float* __restrict__ X, float* __restrict__ T) {
  const int i0 = blockIdx.x * 16, j0 = blockIdx.y * 16;
  const int L = threadIdx.x, half = L >> 4, lm = L & 15;
  v8f acc = {};
#pragma unroll 4
  for (int k = 0; k < 256; k += 4) {
    const int ka = k + 2 * half;
    v2f a, b;
    a.x = X[ka * 256 + i0 + lm];
    a.y = X[(ka + 1) * 256 + i0 + lm];
    b.x = X[ka * 256 + j0 + lm];
    b.y = X[(ka + 1) * 256 + j0 + lm];
    acc = wmma_f32x4(a, b, acc);
  }
#pragma unroll
  for (int r = 0; r < 8; ++r)
    T[(i0 + r + 8 * half) * 256 + j0 + lm] = acc[r];
}

// ---------------------------------------------------------------------------
// Y = 1.5*X - 0.5 * X*T   (one wave per 16x16 tile)
// ---------------------------------------------------------------------------
__global__ void k_nsupdate(const float* __restrict__ X,
                           const float* __restrict__ T,
                           float* __restrict__ Y) {
  const int c0 = blockIdx.x * 16, o0 = blockIdx.y * 16;
  const int L = threadIdx.x, half = L >> 4, lm = L & 15;
  v8f acc = {};
#pragma unroll 4
  for (int k = 0; k < 256; k += 4) {
    const float* ap = &X[(c0 + lm) * 256 + k + 2 * half];
    const int kb = k + 2 * half;
    v2f a, b;
    a.x = ap[0];            // contiguous pair -> b64
    a.y = ap[1];
    b.x = T[kb * 256 + o0 + lm];
    b.y = T[(kb + 1) * 256 + o0 + lm];
    acc = wmma_f32x4(a, b, acc);
  }
#pragma unroll
  for (int r = 0; r < 8; ++r) {
    const int c = c0 + r + 8 * half;
    Y[c * 256 + o0 + lm] = 1.5f * X[c * 256 + o0 + lm] - 0.5f * acc[r];
  }
}

// ---------------------------------------------------------------------------
// Pack orthogonalized weight into K-pair-interleaved layout:
//   P[k2*256 + m] = { W[2k2][m], W[2k2+1][m] }   (k2 = 0..127)
// so the conv's A operand is a single global_load_b64.
// ---------------------------------------------------------------------------
__global__ void k_pack(const float* __restrict__ X, float2* __restrict__ P) {
  const int i = blockIdx.x * 256 + threadIdx.x;  // 32768 pairs
  const int k2 = i >> 8, m = i & 255;
  P[k2 * 256 + m] = make_float2(X[(2 * k2) * 256 + m],
                                X[(2 * k2 + 1) * 256 + m]);
}

// ---------------------------------------------------------------------------
// Main conv:  out[b,o,h,w] = sum_c x[b,c,2h,2w] * Wo[c,o] + bias[o]
// Workgroup (128 thr = 4 waves) per (b, h, 32-wide w-strip); strips at
// w0 = 0 and 24 (overlap idempotent). x tile (256 ch x 32 pos) staged to LDS
// once as float2 K-pairs. Each wave: 4 M-tiles x 2 N-tiles = 8 WMMA per
// k-step fed by 4 global b64 (A, L2-hot) + 2 ds b64 (B).
// ---------------------------------------------------------------------------
__global__ void __launch_bounds__(128)
k_conv(const float* __restrict__ x, const float2* __restrict__ wp,
       const float* __restrict__ bias, float* __restrict__ out) {
  const int tid   = blockIdx.x;
  const int strip = tid & 1;
  const int h     = (tid >> 1) % H_OUT;
  const int b     = tid / (2 * H_OUT);
  const int w0    = strip ? (W_OUT - 32) : 0;   // 0 or 24

  __shared__ float2 lds2[128 * LDSW];           // 48 KB

  const int t = threadIdx.x;
  // ---- stage x tile: lds2[k2][n] = { x[2k2, pos n], x[2k2+1, pos n] } ----
  {
    const int n  = t & 31;
    const int c4 = t >> 5;                      // 0..3
    const size_t base = (size_t)b * C_IN * H_IN * W_IN +
                        (size_t)(2 * h) * W_IN + (size_t)(2 * (w0 + n));
#pragma unroll
    for (int i = 0; i < 32; ++i) {
      const int k2 = c4 + (i << 2);             // 0..127
      const float v0 = x[base + (size_t)(2 * k2) * (H_IN * W_IN)];
      const float v1 = x[base + (size_t)(2 * k2 + 1) * (H_IN * W_IN)];
      lds2[k2 * LDSW + n] = make_float2(v0, v1);
    }
  }
  __syncthreads();

  const int wave = t >> 5;                      // 0..3
  const int L = t & 31, half = L >> 4, lm = L & 15;
  const int m0 = wave * 64;

  v8f acc[4][2] = {{v8f{}, v8f{}}, {v8f{}, v8f{}},
                   {v8f{}, v8f{}}, {v8f{}, v8f{}}};

#pragma unroll 2
  for (int k = 0; k < C_IN; k += 4) {
    const int k2 = (k >> 1) + half;             // pair row for this half-wave
    const float2 b0 = lds2[k2 * LDSW + lm];
    const float2 b1 = lds2[k2 * LDSW + 16 + lm];
    v2f bm0, bm1;
    bm0.x = b0.x; bm0.y = b0.y;
    bm1.x = b1.x; bm1.y = b1.y;
    const float2* wrow = wp + (size_t)k2 * C_OUT + lm;
#pragma unroll
    for (int j = 0; j < 4; ++j) {
      const float2 ap = wrow[m0 + 16 * j];      // single b64, L2-resident
      v2f a;
      a.x = ap.x; a.y = ap.y;
      acc[j][0] = wmma_f32x4(a, bm0, acc[j][0]);
      acc[j][1] = wmma_f32x4(a, bm1, acc[j][1]);
    }
  }

  // ---- epilogue: D row r -> o = m0+16j + r + 8*half, col = w0 + 16*nt + lm
#pragma unroll
  for (int j = 0; j < 4; ++j) {
#pragma unroll
    for (int nt = 0; nt < 2; ++nt) {
#pragma unroll
      for (int r = 0; r < 8; ++r) {
        const int o = m0 + 16 * j + r + 8 * half;
        const size_t oidx = (((size_t)b * C_OUT + o) * H_OUT + h) * W_OUT +
                            (w0 + 16 * nt + lm);
        out[oidx] = acc[j][nt][r] + bias[o];
      }
    }
  }
}

// ---------------------------------------------------------------------------
extern "C" void kernel_launch(void* const* d_in, const int* in_sizes, int n_in,
                              void* d_out, int out_size, void* d_ws,
                              size_t ws_size, hipStream_t stream) {
  const float* d_x    = (const float*)d_in[0];  // (32,256,112,112)
  const float* d_w    = (const float*)d_in[1];  // (256,256)
  const float* d_bias = (const float*)d_in[2];  // (256,1,1)
  float* out = (float*)d_out;

  // workspace layout: X | Y | T/packed | norm2   (~768 KB + 4 B)
  float* bufA  = (float*)d_ws;
  float* bufB  = bufA + 256 * 256;
  float* bufT  = bufB + 256 * 256;
  float* norm2 = bufT + 256 * 256;

  // 1) X0 = W / ||W||_F
  k_zero<<<1, 1, 0, stream>>>(norm2);
  k_frob<<<256, 256, 0, stream>>>(d_w, norm2);
  k_scale<<<256, 256, 0, stream>>>(d_w, norm2, bufA);

  // 2) Newton-Schulz polar iteration: X <- 1.5X - 0.5*X*(X^T X)
  float* X = bufA;
  float* Y = bufB;
  dim3 tiles(16, 16);
  for (int it = 0; it < NS_ITERS; ++it) {
    k_gram<<<tiles, 32, 0, stream>>>(X, bufT);
    k_nsupdate<<<tiles, 32, 0, stream>>>(X, bufT, Y);
    float* tmp = X; X = Y; Y = tmp;
  }

  // 3) pack weight K-pairs (reuses the Gram buffer), then strided conv
  float2* wpacked = (float2*)bufT;
  k_pack<<<128, 256, 0, stream>>>(X, wpacked);

  const int grid = BATCH * H_OUT * 2;  // (b, h, 2 w-strips of 32)
  k_conv<<<grid, 128, 0, stream>>>(d_x, wpacked, d_bias, out);
}